// L2_LOSS_80951543595378
// MI455X (gfx1250) — compile-verified
//
#include <hip/hip_runtime.h>
#include <cstdint>

// ---------------------------------------------------------------------------
// FK loss kernel for MI455X (gfx1250).
// Memory-bound: 377 MB streamed @ 23.3 TB/s => ~16us floor; ~1 GFLOP fp32.
// Data movement: CDNA5 Tensor Data Mover (tensor_load_to_lds) stages 23KB
// tiles of pred/y into LDS (verified: the 4 "other" ops in round-2 histogram).
// Compute: per-thread fp32 VALU (3x3 chains have no 16x16xK WMMA structure).
// Reductions: wave32 sum via V_WMMA_F32_16X16X4_F32 ones-matrix trick.
// ---------------------------------------------------------------------------

typedef unsigned int u32x4 __attribute__((ext_vector_type(4)));
typedef int          i32x8 __attribute__((ext_vector_type(8)));
typedef int          i32x4 __attribute__((ext_vector_type(4)));
typedef float        v2f   __attribute__((ext_vector_type(2)));
typedef float        v8f   __attribute__((ext_vector_type(8)));

#define ROWS_PER_BLOCK 64
#define FPR            90                      // floats per sample row
#define TILE_FLOATS    (ROWS_PER_BLOCK * FPR)  // 5760
#define TILE_BYTES     (TILE_FLOATS * 4)       // 23040

// 1-D TDM load: `elems` consecutive f32 from global -> LDS byte offset.
// D# layout per CDNA5 ISA (2D descriptor, groups
// 2/3 zero). Issued once per wave; EXEC is ignored by TDM.
// This toolchain's builtin is the 6-arg form:
//   (u32x4 g0, i32x8 g1, i32x4 g2, i32x4 g3, i32x8 extra, i32 cpol)
__device__ __forceinline__ void tdm_load_1d(const void* gptr, unsigned ldsByteOff,
                                            unsigned elems)
{
  unsigned long long ga = (unsigned long long)(uintptr_t)gptr;
  u32x4 g0 = { 1u,                                          // count=1, user desc
               ldsByteOff,                                  // lds_addr [63:32]
               (unsigned)(ga & 0xFFFFFFFFull),              // global_addr[31:0]
               (unsigned)((ga >> 32) & 0x01FFFFFFull)       // global_addr[56:32]
                 | 0x80000000u };                           // type=2 ("image")
  i32x8 g1 = { (int)(2u << 16),                             // data_size=4B
               (int)((elems & 0xFFFFu) << 16),              // tensor_dim0[15:0]
               (int)(((elems >> 16) & 0xFFFFu) | (1u<<16)), // dim0[31:16], tensor_dim1=1
               (int)((elems & 0xFFFFu) << 16),              // tile_dim0 = elems (<=5760)
               1,                                           // tile_dim1=1, tile_dim2=0
               (int)elems,                                  // tensor_dim0_stride lo
               0, 0 };
  i32x4 z4 = { 0, 0, 0, 0 };
  i32x8 z8 = { 0, 0, 0, 0, 0, 0, 0, 0 };
  __builtin_amdgcn_tensor_load_to_lds(g0, g1, z4, z4, z8, 0);
}

// Wave32 sum via one V_WMMA_F32_16X16X4_F32 (ones-matrix trick).
// A slots all = lane value v (each lane owns exactly 2 of A's 64 slots),
// B = ones  =>  D[m][n] = 2*(v_m + v_{m+16}) for every n.
// Sum of a lane's 8 D registers = 2*sum(rows 0-7) (lanes 0-15) or
// 2*sum(rows 8-15) (lanes 16-31); combine halves with shfl_xor(16), halve.
// Requires EXEC all-1s in the wave (all call sites are non-divergent).
__device__ __forceinline__ float wave_sum_wmma(float v)
{
  v2f a = { v, v };
  v2f b = { 1.0f, 1.0f };
  v8f c = { 0.0f, 0.0f, 0.0f, 0.0f, 0.0f, 0.0f, 0.0f, 0.0f };
  c = __builtin_amdgcn_wmma_f32_16x16x4_f32(false, a, false, b,
                                            (short)0, c, false, false);
  float s = ((c[0] + c[1]) + (c[2] + c[3])) + ((c[4] + c[5]) + (c[6] + c[7]));
  s += __shfl_xor(s, 16, 32);
  return 0.5f * s;
}

struct Frame { float R[9]; float p[3]; float t[3]; };

// 6D -> rotation matrix, columns = [b1 b2 b3] (matches jnp.stack(..., axis=-1)).
__device__ __forceinline__ void r6d_to_R(const float* __restrict__ r, float R[9])
{
  float a1x=r[0], a1y=r[1], a1z=r[2];
  float a2x=r[3], a2y=r[4], a2z=r[5];
  float n1 = rsqrtf(a1x*a1x + a1y*a1y + a1z*a1z);
  float b1x=a1x*n1, b1y=a1y*n1, b1z=a1z*n1;
  float d  = b1x*a2x + b1y*a2y + b1z*a2z;
  float c2x=a2x-d*b1x, c2y=a2y-d*b1y, c2z=a2z-d*b1z;
  float n2 = rsqrtf(c2x*c2x + c2y*c2y + c2z*c2z);
  float b2x=c2x*n2, b2y=c2y*n2, b2z=c2z*n2;
  float b3x=b1y*b2z-b1z*b2y;
  float b3y=b1z*b2x-b1x*b2z;
  float b3z=b1x*b2y-b1y*b2x;
  R[0]=b1x; R[1]=b2x; R[2]=b3x;
  R[3]=b1y; R[4]=b2y; R[5]=b3y;
  R[6]=b1z; R[7]=b2z; R[8]=b3z;
}

__device__ __forceinline__ void mul33(const float A[9], const float Bm[9], float C[9])
{
#pragma unroll
  for (int r = 0; r < 3; ++r)
#pragma unroll
    for (int c = 0; c < 3; ++c)
      C[r*3+c] = A[r*3+0]*Bm[0*3+c] + A[r*3+1]*Bm[1*3+c] + A[r*3+2]*Bm[2*3+c];
}

// wave32 WMMA-reduce then one ds_add_f32 per wave per joint.
__device__ __forceinline__ void accum(float* __restrict__ acc, int j, float e)
{
  float s = wave_sum_wmma(e);
  if ((threadIdx.x & 31) == 0)
    __hip_atomic_fetch_add(&acc[j], s, __ATOMIC_RELAXED, __HIP_MEMORY_SCOPE_WORKGROUP);
}

// Interior joint: position via parent frame, then update frame rotation.
template<int J>
__device__ __forceinline__ void advance(Frame& f, const float* __restrict__ P,
                                        const float* __restrict__ T,
                                        const float* __restrict__ LOC,
                                        float* __restrict__ acc, float valid)
{
  const float tj0 = T[3*J+0], tj1 = T[3*J+1], tj2 = T[3*J+2];
  const float ox = tj0 - f.t[0], oy = tj1 - f.t[1], oz = tj2 - f.t[2];
  const float p0 = f.p[0] + f.R[0]*ox + f.R[1]*oy + f.R[2]*oz;
  const float p1 = f.p[1] + f.R[3]*ox + f.R[4]*oy + f.R[5]*oz;
  const float p2 = f.p[2] + f.R[6]*ox + f.R[7]*oy + f.R[8]*oz;
  const float dx = p0 - LOC[3*J+0], dy = p1 - LOC[3*J+1], dz = p2 - LOC[3*J+2];
  accum(acc, J, valid * sqrtf(dx*dx + dy*dy + dz*dz));
  float Rl[9]; r6d_to_R(P + 6*J, Rl);
  float Rn[9]; mul33(f.R, Rl, Rn);
#pragma unroll
  for (int k = 0; k < 9; ++k) f.R[k] = Rn[k];
  f.p[0]=p0; f.p[1]=p1; f.p[2]=p2;
  f.t[0]=tj0; f.t[1]=tj1; f.t[2]=tj2;
}

// Leaf joint: its own rotation never contributes to the loss -> skip it.
template<int J>
__device__ __forceinline__ void leaf(const Frame& f,
                                     const float* __restrict__ T,
                                     const float* __restrict__ LOC,
                                     float* __restrict__ acc, float valid)
{
  const float ox = T[3*J+0] - f.t[0], oy = T[3*J+1] - f.t[1], oz = T[3*J+2] - f.t[2];
  const float p0 = f.p[0] + f.R[0]*ox + f.R[1]*oy + f.R[2]*oz;
  const float p1 = f.p[1] + f.R[3]*ox + f.R[4]*oy + f.R[5]*oz;
  const float p2 = f.p[2] + f.R[6]*ox + f.R[7]*oy + f.R[8]*oz;
  const float dx = p0 - LOC[3*J+0], dy = p1 - LOC[3*J+1], dz = p2 - LOC[3*J+2];
  accum(acc, J, valid * sqrtf(dx*dx + dy*dy + dz*dz));
}

template<bool USE_ATOMIC>
__global__ void __launch_bounds__(ROWS_PER_BLOCK)
fk_main(const float* __restrict__ pred, const float* __restrict__ y,
        float* __restrict__ dst, int B, int nb, float invB)
{
  extern __shared__ float smem[];            // [predTile | yTile | acc(16)]
  float* predS = smem;
  float* yS    = smem + TILE_FLOATS;
  float* acc   = smem + 2 * TILE_FLOATS;

  const int tid  = (int)threadIdx.x;
  const int base = (int)blockIdx.x * ROWS_PER_BLOCK;
  int rows = B - base; if (rows > ROWS_PER_BLOCK) rows = ROWS_PER_BLOCK;
  const unsigned elems = (unsigned)rows * FPR;

  if (tid < 16) acc[tid] = 0.0f;
  if (tid < 32) {                            // wave 0 drives the TDM
    tdm_load_1d(pred + (size_t)base * FPR, 0u, elems);
    tdm_load_1d(y    + (size_t)base * FPR, (unsigned)TILE_BYTES, elems);
    __builtin_amdgcn_s_wait_tensorcnt(0);
  }
  __syncthreads();

  const int   tr    = (tid < rows) ? tid : 0;     // clamp so reductions see real data
  const float valid = (tid < rows) ? 1.0f : 0.0f;
  const float* __restrict__ P   = predS + tr * FPR;   // 15 x 6 r6d
  const float* __restrict__ LOC = yS    + tr * FPR;   // y[:, 0:45]
  const float* __restrict__ T   = LOC + 45;           // y[:, 45:90]

  // Root joint 0.
  Frame f0;
  r6d_to_R(P, f0.R);
  f0.t[0]=T[0]; f0.t[1]=T[1]; f0.t[2]=T[2];
  f0.p[0]=f0.t[0]; f0.p[1]=f0.t[1]; f0.p[2]=f0.t[2];
  {
    const float dx=f0.p[0]-LOC[0], dy=f0.p[1]-LOC[1], dz=f0.p[2]-LOC[2];
    accum(acc, 0, valid * sqrtf(dx*dx + dy*dy + dz*dz));
  }

  // PARENTS = [-1,0,0,0,1,2,3,4,5,6,7,8,9,9,9] -> three chains off the root.
  Frame f = f0;
  advance<1>(f,P,T,LOC,acc,valid); advance<4>(f,P,T,LOC,acc,valid);
  advance<7>(f,P,T,LOC,acc,valid); leaf<10>(f,T,LOC,acc,valid);
  f = f0;
  advance<2>(f,P,T,LOC,acc,valid); advance<5>(f,P,T,LOC,acc,valid);
  advance<8>(f,P,T,LOC,acc,valid); leaf<11>(f,T,LOC,acc,valid);
  f = f0;
  advance<3>(f,P,T,LOC,acc,valid); advance<6>(f,P,T,LOC,acc,valid);
  advance<9>(f,P,T,LOC,acc,valid);
  leaf<12>(f,T,LOC,acc,valid); leaf<13>(f,T,LOC,acc,valid); leaf<14>(f,T,LOC,acc,valid);

  __syncthreads();
  if (tid < 15) {
    if (USE_ATOMIC)
      __hip_atomic_fetch_add(&dst[tid], acc[tid] * invB,
                             __ATOMIC_RELAXED, __HIP_MEMORY_SCOPE_AGENT);
    else
      dst[(size_t)tid * nb + blockIdx.x] = acc[tid];   // atomic-free partials
  }
}

__global__ void __launch_bounds__(256)
fk_reduce(const float* __restrict__ ws, float* __restrict__ out, int nb, float invB)
{
  __shared__ float red[8];
  const int j = (int)blockIdx.x;          // one block per joint
  float s = 0.0f;
  for (int i = (int)threadIdx.x; i < nb; i += 256)
    s += ws[(size_t)j * nb + i];
  const int lane = (int)threadIdx.x & 31, w = (int)threadIdx.x >> 5;
  s = wave_sum_wmma(s);                   // wave32 sum on the matrix pipe
  if (lane == 0) red[w] = s;
  __syncthreads();
  if (w == 0) {                           // whole wave active -> EXEC all-1s
    s = (lane < 8) ? red[lane] : 0.0f;
    s = wave_sum_wmma(s);
    if (lane == 0) out[j] = s * invB;
  }
}

extern "C" void kernel_launch(void* const* d_in, const int* in_sizes, int n_in,
                              void* d_out, int out_size, void* d_ws, size_t ws_size,
                              hipStream_t stream)
{
  const float* pred = (const float*)d_in[0];
  const float* y    = (const float*)d_in[1];
  float* out        = (float*)d_out;

  const long long B = (long long)in_sizes[0] / FPR;
  if (B <= 0) return;
  const int   nb   = (int)((B + ROWS_PER_BLOCK - 1) / ROWS_PER_BLOCK);
  const float invB = 1.0f / (float)B;
  const size_t shmem = (size_t)(2 * TILE_FLOATS + 16) * sizeof(float); // ~46.1 KB
  const size_t need  = (size_t)nb * 15 * sizeof(float);

  if (ws_size >= need) {
    fk_main<false><<<nb, ROWS_PER_BLOCK, shmem, stream>>>(pred, y, (float*)d_ws,
                                                          (int)B, nb, invB);
    fk_reduce<<<15, 256, 0, stream>>>((const float*)d_ws, out, nb, invB);
  } else {
    (void)hipMemsetAsync(out, 0, 15 * sizeof(float), stream);
    fk_main<true><<<nb, ROWS_PER_BLOCK, shmem, stream>>>(pred, y, out,
                                                         (int)B, nb, invB);
  }
}